// GAT_76682346102828
// MI455X (gfx1250) — compile-verified
//
#include <hip/hip_runtime.h>
#include <math.h>

// ---------------- problem constants (from reference) ----------------
#define N_PAPER   30000
#define N_AUTHOR  20000
#define N_NODES   50000
#define N_EDGES   800000
#define D_PAPER   512
#define D_AUTHOR  256
#define N_HID     128
#define N_OUT     64
#define HEADS1    2
#define HEADS2    1
#define NEG_SLOPE 0.2f

typedef __attribute__((ext_vector_type(2))) float v2f;
typedef __attribute__((ext_vector_type(8))) float v8f;

#define JT 4   // j-tiles per wave: wave computes a 16 x (16*JT) = 16x64 strip

// ---------------------------------------------------------------
// fp32 WMMA GEMM:  C[M,N] = act(A[M,K] @ B[K,N] + bias)
// One wave32 computes a 16x64 strip via 4 accumulators of
// V_WMMA_F32_16X16X4_F32, reusing the A fragment across the 4 j-tiles.
// A fragment (16x4 fp32, 2 VGPRs): lane l(0..15)/hi(0..1): K = 2*hi + v.
// B fragment (4x16 fp32, 2 VGPRs): symmetric, N = lane&15.
// C/D (8 VGPRs): VGPR r -> row i0 + r + 8*hi, col j0 + (lane&15).
// Requires M % 16 == 0, N % 64 == 0, K % 64 == 0 (true for all GEMMs here).
// act: 0 = none, 1 = relu
// ---------------------------------------------------------------
__global__ void gat_wmma_gemm_f32(const float* __restrict__ A,
                                  const float* __restrict__ B,
                                  const float* __restrict__ bias,
                                  float* __restrict__ C,
                                  int M, int K, int N, int act) {
    const int lane = threadIdx.x & 31;
    const int nsj  = N >> 6;                         // 64-wide strips along N
    const int numStrips = (M >> 4) * nsj;
    const int strip = blockIdx.x * (blockDim.x >> 5) + (threadIdx.x >> 5);
    if (strip >= numStrips) return;                  // whole-wave exit

    const int ti = strip / nsj;
    const int tj = strip - ti * nsj;
    const int i0 = ti << 4, j0 = tj << 6;
    const int l  = lane & 15;
    const int hi = lane >> 4;

    v8f acc[JT] = {};
    const float* __restrict__ arow = A + (size_t)(i0 + l) * K;

    // K is a multiple of 64: outer chunk loop hoists the A-stream prefetch
    // out of the hot body (no per-iteration guard SALU).
    for (int k0 = 0; k0 < K; k0 += 64) {
        if (k0 + 64 < K)
            __builtin_prefetch(arow + k0 + 64, 0, 3);   // near-scope prefetch

#pragma unroll 4
        for (int kk = 0; kk < 64; kk += 4) {
            const int ka = k0 + kk + 2 * hi;
            v2f a;
            a.x = arow[ka];
            a.y = arow[ka + 1];
            const float* __restrict__ b0 = B + (size_t)ka * N + j0 + l;
            const float* __restrict__ b1 = b0 + N;
#pragma unroll
            for (int t = 0; t < JT; ++t) {
                v2f b;
                b.x = b0[16 * t];
                b.y = b1[16 * t];
                acc[t] = __builtin_amdgcn_wmma_f32_16x16x4_f32(
                             /*neg_a=*/false, a, /*neg_b=*/false, b,
                             /*c_mod=*/(short)0, acc[t],
                             /*reuse_a=*/false, /*reuse_b=*/false);
            }
        }
    }

#pragma unroll
    for (int t = 0; t < JT; ++t) {
        const int col = j0 + 16 * t + l;
        const float bv = bias ? bias[col] : 0.0f;
#pragma unroll
        for (int r = 0; r < 8; ++r) {
            float v = acc[t][r] + bv;
            if (act == 1) v = fmaxf(v, 0.0f);
            C[(size_t)(i0 + r + 8 * hi) * N + col] = v;
        }
    }
}

// ---------------- fill helper ----------------
__global__ void gat_fill_f32(float* __restrict__ p, float v, int n) {
    for (int i = blockIdx.x * blockDim.x + threadIdx.x; i < n;
         i += gridDim.x * blockDim.x)
        p[i] = v;
}

// ---------------- per-node attention logits: as/ad[n*H+h] = <h[n,h,:], a> ----
__global__ void gat_alpha(const float* __restrict__ h,
                          const float* __restrict__ a_src,
                          const float* __restrict__ a_dst,
                          float* __restrict__ as, float* __restrict__ ad,
                          int n_nodes, int H, int C) {
    const int warp = blockIdx.x * (blockDim.x >> 5) + (threadIdx.x >> 5);
    const int lane = threadIdx.x & 31;
    const int total = n_nodes * H;
    if (warp >= total) return;
    const int n = warp / H, hh = warp - n * H;
    const float* __restrict__ hp = h + (size_t)n * H * C + (size_t)hh * C;
    float ss = 0.0f, sd = 0.0f;
    for (int cc = lane; cc < C; cc += 32) {
        const float v = hp[cc];
        ss += v * a_src[hh * C + cc];
        sd += v * a_dst[hh * C + cc];
    }
#pragma unroll
    for (int off = 16; off; off >>= 1) {
        ss += __shfl_xor(ss, off, 32);
        sd += __shfl_xor(sd, off, 32);
    }
    if (lane == 0) { as[warp] = ss; ad[warp] = sd; }
}

// float atomic max via signed/unsigned int ordering trick (init must be -inf)
__device__ __forceinline__ void atomicMaxF32(float* addr, float val) {
    if (val >= 0.0f) atomicMax((int*)addr, __float_as_int(val));
    else             atomicMin((unsigned int*)addr, __float_as_uint(val));
}

__device__ __forceinline__ float leaky(float v) {
    return v > 0.0f ? v : NEG_SLOPE * v;
}

// ---------------- edge pass A: segment max of leaky_relu logits --------------
__global__ void gat_edge_max(const int* __restrict__ src,
                             const int* __restrict__ dst,
                             const float* __restrict__ as,
                             const float* __restrict__ ad,
                             float* __restrict__ m, int nE, int H) {
    const int total = nE * H;
    for (int i = blockIdx.x * blockDim.x + threadIdx.x; i < total;
         i += gridDim.x * blockDim.x) {
        const int e = i / H, hh = i - e * H;
        const int s = src[e], d = dst[e];
        atomicMaxF32(&m[d * H + hh], leaky(as[s * H + hh] + ad[d * H + hh]));
    }
}

// ---------------- edge pass B: numerator / denominator accumulation --------
// wave per edge: lanes stride the C channels; lane 0 owns the denominator add
__global__ void gat_edge_acc(const int* __restrict__ src,
                             const int* __restrict__ dst,
                             const float* __restrict__ as,
                             const float* __restrict__ ad,
                             const float* __restrict__ m,
                             const float* __restrict__ h,
                             float* __restrict__ den,
                             float* __restrict__ out,
                             int nE, int H, int C) {
    const int warp   = blockIdx.x * (blockDim.x >> 5) + (threadIdx.x >> 5);
    const int lane   = threadIdx.x & 31;
    const int nwarps = gridDim.x * (blockDim.x >> 5);
    const int HC = H * C;
    for (int e = warp; e < nE; e += nwarps) {
        const int s = src[e], d = dst[e];
        for (int hh = 0; hh < H; ++hh) {
            const float logit = leaky(as[s * H + hh] + ad[d * H + hh]);
            const float w = __expf(logit - m[d * H + hh]);
            if (lane == 0) atomicAdd(&den[d * H + hh], w);
            const float* __restrict__ hs = h   + (size_t)s * HC + hh * C;
            float*       __restrict__ od = out + (size_t)d * HC + hh * C;
            for (int cc = lane; cc < C; cc += 32)
                atomicAdd(&od[cc], w * hs[cc]);
        }
    }
}

// ---------------- finalize layer1: x = elu(acc/den + bias), in place --------
__global__ void gat_finalize_elu(float* __restrict__ out,
                                 const float* __restrict__ den,
                                 const float* __restrict__ bias,
                                 int n_nodes, int H, int C) {
    const int HC = H * C;
    const int total = n_nodes * HC;
    for (int i = blockIdx.x * blockDim.x + threadIdx.x; i < total;
         i += gridDim.x * blockDim.x) {
        const int n = i / HC, r = i - n * HC;   // r = hh*C + cc
        const int hh = r / C;
        float v = out[i] / (den[n * H + hh] + 1e-16f) + bias[r];
        out[i] = v > 0.0f ? v : __expf(v) - 1.0f;   // ELU(alpha=1)
    }
}

// ---------------- finalize layer2: out = acc/den + bias (1 head mean) --------
__global__ void gat_finalize_out(float* __restrict__ out,
                                 const float* __restrict__ den,
                                 const float* __restrict__ bias,
                                 int n_nodes, int C) {
    const int total = n_nodes * C;
    for (int i = blockIdx.x * blockDim.x + threadIdx.x; i < total;
         i += gridDim.x * blockDim.x) {
        const int n = i / C, cc = i - n * C;
        out[i] = out[i] / (den[n] + 1e-16f) + bias[cc];
    }
}

// =====================================================================
extern "C" void kernel_launch(void* const* d_in, const int* in_sizes, int n_in,
                              void* d_out, int out_size, void* d_ws, size_t ws_size,
                              hipStream_t stream) {
    const float* x_paper = (const float*)d_in[0];
    const float* x_author= (const float*)d_in[1];
    const int*   edge    = (const int*)  d_in[2];
    const float* W1p     = (const float*)d_in[3];
    const float* b1p     = (const float*)d_in[4];
    const float* W1a     = (const float*)d_in[5];
    const float* b1a     = (const float*)d_in[6];
    const float* Wg1     = (const float*)d_in[7];
    const float* asrc1   = (const float*)d_in[8];
    const float* adst1   = (const float*)d_in[9];
    const float* bg1     = (const float*)d_in[10];
    const float* Wg2     = (const float*)d_in[11];
    const float* asrc2   = (const float*)d_in[12];
    const float* adst2   = (const float*)d_in[13];
    const float* bg2     = (const float*)d_in[14];
    float* out = (float*)d_out;

    const int* src = edge;            // edge_index[0]
    const int* dst = edge + N_EDGES;  // edge_index[1]

    // ---- workspace carve-up (256B aligned) ----
    char* base = (char*)d_ws;
    size_t off = 0;
    auto carve = [&](size_t bytes) -> float* {
        float* p = (float*)(base + off);
        off += (bytes + 255) & ~(size_t)255;
        return p;
    };
    float* x    = carve((size_t)N_NODES * N_HID * 4);              // encoded feats [N,128]
    float* h1   = carve((size_t)N_NODES * HEADS1 * N_HID * 4);     // [N,256]
    float* as1  = carve((size_t)N_NODES * HEADS1 * 4);
    float* ad1  = carve((size_t)N_NODES * HEADS1 * 4);
    float* m1   = carve((size_t)N_NODES * HEADS1 * 4);
    float* den1 = carve((size_t)N_NODES * HEADS1 * 4);
    float* out1 = carve((size_t)N_NODES * HEADS1 * N_HID * 4);     // [N,256] -> x2 in place
    float* h2   = carve((size_t)N_NODES * N_OUT * 4);              // [N,64]
    float* as2  = carve((size_t)N_NODES * 4);
    float* ad2  = carve((size_t)N_NODES * 4);
    float* m2   = carve((size_t)N_NODES * 4);
    float* den2 = carve((size_t)N_NODES * 4);
    (void)ws_size; (void)in_sizes; (void)n_in; (void)out_size;

    const int BT = 256;
    auto cdiv = [](long long a, long long b) { return (int)((a + b - 1) / b); };
    const float NINF = -__builtin_huge_valf();

    // ---- init accumulators ----
    gat_fill_f32<<<cdiv(N_NODES*HEADS1, BT), BT, 0, stream>>>(m1,  NINF, N_NODES*HEADS1);
    gat_fill_f32<<<cdiv(N_NODES*HEADS1, BT), BT, 0, stream>>>(den1, 0.f, N_NODES*HEADS1);
    gat_fill_f32<<<cdiv((long long)N_NODES*HEADS1*N_HID, BT), BT, 0, stream>>>(
        out1, 0.f, N_NODES*HEADS1*N_HID);
    gat_fill_f32<<<cdiv(N_NODES, BT), BT, 0, stream>>>(m2,  NINF, N_NODES);
    gat_fill_f32<<<cdiv(N_NODES, BT), BT, 0, stream>>>(den2, 0.f, N_NODES);
    gat_fill_f32<<<cdiv((long long)N_NODES*N_OUT, BT), BT, 0, stream>>>(
        out, 0.f, N_NODES*N_OUT);

    // ---- node-type encoders (WMMA GEMM, fused bias + ReLU) ----
    {
        int strips = (N_PAPER/16) * (N_HID/64);
        gat_wmma_gemm_f32<<<cdiv(strips, 8), BT, 0, stream>>>(
            x_paper, W1p, b1p, x, N_PAPER, D_PAPER, N_HID, 1);
    }
    {
        int strips = (N_AUTHOR/16) * (N_HID/64);
        gat_wmma_gemm_f32<<<cdiv(strips, 8), BT, 0, stream>>>(
            x_author, W1a, b1a, x + (size_t)N_PAPER * N_HID,
            N_AUTHOR, D_AUTHOR, N_HID, 1);
    }

    // ---- GAT layer 1 ----
    {
        int strips = (N_NODES/16) * ((HEADS1*N_HID)/64);
        gat_wmma_gemm_f32<<<cdiv(strips, 8), BT, 0, stream>>>(
            x, Wg1, nullptr, h1, N_NODES, N_HID, HEADS1*N_HID, 0);
    }
    gat_alpha<<<cdiv(N_NODES*HEADS1, 8), BT, 0, stream>>>(
        h1, asrc1, adst1, as1, ad1, N_NODES, HEADS1, N_HID);
    gat_edge_max<<<cdiv((long long)N_EDGES*HEADS1, BT), BT, 0, stream>>>(
        src, dst, as1, ad1, m1, N_EDGES, HEADS1);
    gat_edge_acc<<<cdiv(N_EDGES, 8), BT, 0, stream>>>(
        src, dst, as1, ad1, m1, h1, den1, out1, N_EDGES, HEADS1, N_HID);
    gat_finalize_elu<<<cdiv((long long)N_NODES*HEADS1*N_HID, BT), BT, 0, stream>>>(
        out1, den1, bg1, N_NODES, HEADS1, N_HID);

    // ---- GAT layer 2 ----
    {
        int strips = (N_NODES/16) * (N_OUT/64);
        gat_wmma_gemm_f32<<<cdiv(strips, 8), BT, 0, stream>>>(
            out1, Wg2, nullptr, h2, N_NODES, HEADS1*N_HID, N_OUT, 0);
    }
    gat_alpha<<<cdiv(N_NODES*HEADS2, 8), BT, 0, stream>>>(
        h2, asrc2, adst2, as2, ad2, N_NODES, HEADS2, N_OUT);
    gat_edge_max<<<cdiv((long long)N_EDGES*HEADS2, BT), BT, 0, stream>>>(
        src, dst, as2, ad2, m2, N_EDGES, HEADS2);
    gat_edge_acc<<<cdiv(N_EDGES, 8), BT, 0, stream>>>(
        src, dst, as2, ad2, m2, h2, den2, out, N_EDGES, HEADS2, N_OUT);
    gat_finalize_out<<<cdiv((long long)N_NODES*N_OUT, BT), BT, 0, stream>>>(
        out, den2, bg2, N_NODES, N_OUT);
}